// GCNLayer_28776280883474
// MI455X (gfx1250) — compile-verified
//
#include <hip/hip_runtime.h>

typedef __attribute__((ext_vector_type(2))) float v2f;
typedef __attribute__((ext_vector_type(8))) float v8f;
typedef __attribute__((ext_vector_type(4))) unsigned int v4u;
typedef __attribute__((ext_vector_type(4))) int v4i;
typedef __attribute__((ext_vector_type(8))) int v8i;

#define DIM 128
#define WS_STRIDE 136   /* 128 + 8 dword pad per row (TDM pad) -> conflict-free B reads */
#define BN_EPS 1e-5f

// ---------------------------------------------------------------- zero init
__global__ void zero_kernel(float* __restrict__ p, long n) {
    long i = (long)blockIdx.x * blockDim.x + threadIdx.x;
    long stride = (long)gridDim.x * blockDim.x;
    for (; i < n; i += stride) p[i] = 0.0f;
}

// ---------------------------------------------------------------- degrees
__global__ void degree_kernel(const int* __restrict__ src, const int* __restrict__ dst,
                              float* __restrict__ outd, float* __restrict__ ind, int E) {
    int e = blockIdx.x * blockDim.x + threadIdx.x;
    if (e >= E) return;
    unsafeAtomicAdd(&outd[src[e]], 1.0f);
    unsafeAtomicAdd(&ind[dst[e]], 1.0f);
}

// ------------------------------------------------- x = feature * out_deg^-0.5
__global__ void scale_src_kernel(const float* __restrict__ feat, const float* __restrict__ outd,
                                 float* __restrict__ x, int n) {
    long idx = (long)blockIdx.x * blockDim.x + threadIdx.x;   // over n*32 float4s
    if (idx >= (long)n * (DIM / 4)) return;
    int row = (int)(idx >> 5);
    float ns = rsqrtf(fmaxf(outd[row], 1.0f));
    float4 f = ((const float4*)feat)[idx];
    f.x *= ns; f.y *= ns; f.z *= ns; f.w *= ns;
    ((float4*)x)[idx] = f;
}

// ------------------------------------------------- agg[dst] += x[src] (scatter)
// one wave per edge, float4 per lane -> coalesced 512B gather + 4 f32 atomics/lane
// (agg is 25.6 MB -> atomics resolve in the 192 MB L2)
__global__ __launch_bounds__(256) void edge_scatter_kernel(const int* __restrict__ src,
                                                           const int* __restrict__ dst,
                                                           const float* __restrict__ x,
                                                           float* __restrict__ agg, int E) {
    int tid = blockIdx.x * blockDim.x + threadIdx.x;
    int e = tid >> 5;
    if (e >= E) return;
    int lane = tid & 31;
    int s = src[e];
    int d = dst[e];
    float4 v = ((const float4*)x)[(long)s * (DIM / 4) + lane];
    float* ap = agg + (long)d * DIM + lane * 4;
    unsafeAtomicAdd(ap + 0, v.x);
    unsafeAtomicAdd(ap + 1, v.y);
    unsafeAtomicAdd(ap + 2, v.z);
    unsafeAtomicAdd(ap + 3, v.w);
}

// ------------------------------------------------- h = (agg * in_deg^-0.5) @ W + b
// WMMA f32 16x16x4: wave computes a 16x128 slab; 4 waves/block share W in LDS,
// staged by the Tensor Data Mover (with row padding to kill LDS bank conflicts).
__global__ __launch_bounds__(128) void gemm_wmma_kernel(const float* __restrict__ agg,
                                                        const float* __restrict__ ind,
                                                        const float* __restrict__ W,
                                                        const float* __restrict__ bias,
                                                        float* __restrict__ h, int n) {
    __shared__ float Ws[DIM * WS_STRIDE];   // ~68 KB of the 320 KB WGP LDS

    const int tid = threadIdx.x;

#if __has_builtin(__builtin_amdgcn_tensor_load_to_lds)
    if (tid < 32) {   // wave 0 drives the TDM (EXEC ignored, per-wave op)
        typedef __attribute__((address_space(3))) float lds_f;
        unsigned ldsoff = (unsigned)(unsigned long long)(lds_f*)Ws;
        unsigned long long ga = (unsigned long long)W;
        // D# group0: count=1 | lds_addr | global_addr[56:0] | type=2
        v4u g0 = { 1u, ldsoff, (unsigned)(ga & 0xFFFFFFFFu),
                   (unsigned)((ga >> 32) & 0x01FFFFFFu) | 0x80000000u };
        // D# group1: data_size=4B, pad_enable, pad_interval=128dw, pad_amount=8dw,
        //            tensor_dim0=128, tensor_dim1=128, tile 128x128, stride0=128
        v8i g1 = { (int)0x0F920000,      // dsz=2 | pad_en | intv=6 | amt=7
                   (int)(128u << 16),    // tensor_dim0[15:0]
                   (int)(128u << 16),    // tensor_dim1[15:0]
                   (int)(128u << 16),    // tile_dim0
                   128,                  // tile_dim1
                   128,                  // tensor_dim0_stride[31:0]
                   0x40000000,           // tensor_dim1_stride[15:0]=16384
                   0 };
        v4i gz4 = { 0, 0, 0, 0 };
#if __clang_major__ >= 23
        v8i gz8 = { 0, 0, 0, 0, 0, 0, 0, 0 };
        __builtin_amdgcn_tensor_load_to_lds(g0, g1, gz4, gz4, gz8, 0);
#else
        __builtin_amdgcn_tensor_load_to_lds(g0, g1, gz4, gz4, 0);
#endif
        __builtin_amdgcn_s_wait_tensorcnt(0);
    }
#else
    // fallback: cooperative staging into the same padded layout
    for (int i = tid; i < DIM * DIM; i += 128) {
        int r = i >> 7, c = i & 127;
        Ws[r * WS_STRIDE + c] = W[i];
    }
#endif
    __syncthreads();

    const int wave = tid >> 5;
    const int lane = tid & 31;
    const int mtile = blockIdx.x * 4 + wave;
    const int mrow_base = mtile * 16;

    const int mr = lane & 15;            // row within tile (A) / column within tile (B,C)
    const int kofs = (lane >> 4) << 1;   // lanes 0-15: K+0/K+1 ; lanes 16-31: K+2/K+3

    int row = mrow_base + mr;
    if (row >= n) row = n - 1;           // clamp tail loads; EXEC stays all-1s for WMMA
    const float* arow = agg + (long)row * DIM;
    const float nd = rsqrtf(fmaxf(ind[row], 1.0f));

    v8f acc[8] = {};

    for (int kk = 0; kk < DIM; kk += 4) {
        v2f a;
        a.x = arow[kk + kofs] * nd;
        a.y = arow[kk + kofs + 1] * nd;
#pragma unroll
        for (int t = 0; t < 8; ++t) {
            v2f bfr;
            const int col = t * 16 + mr;
            bfr.x = Ws[(kk + kofs) * WS_STRIDE + col];
            bfr.y = Ws[(kk + kofs + 1) * WS_STRIDE + col];
            acc[t] = __builtin_amdgcn_wmma_f32_16x16x4_f32(
                false, a, false, bfr, (short)0, acc[t], false, false);
        }
    }

    // C/D layout: VGPR v -> M=v (lanes 0-15) or M=v+8 (lanes 16-31); N = lane&15
    const int rbase = mrow_base + ((lane >> 4) << 3);
    if (mrow_base + 16 <= n) {
        // fast path: whole tile in range, unguarded stores
#pragma unroll
        for (int t = 0; t < 8; ++t) {
            const int c = t * 16 + mr;
            const float bb = bias[c];
#pragma unroll
            for (int v = 0; v < 8; ++v) {
                h[(long)(rbase + v) * DIM + c] = acc[t][v] + bb;
            }
        }
    } else {
        // tail tile: per-row guard
#pragma unroll
        for (int t = 0; t < 8; ++t) {
            const int c = t * 16 + mr;
            const float bb = bias[c];
#pragma unroll
            for (int v = 0; v < 8; ++v) {
                const int r = rbase + v;
                if (r < n) h[(long)r * DIM + c] = acc[t][v] + bb;
            }
        }
    }
}

// ------------------------------------------------- per-channel sum / sumsq
// grid stride is a multiple of 128 -> each thread's channel is fixed; register
// accumulate, fold the two half-blocks in LDS, one atomic per channel per block.
__global__ __launch_bounds__(256) void bn_stats_kernel(const float* __restrict__ h,
                                                       float* __restrict__ stats, long total) {
    __shared__ float ssum[256];
    __shared__ float ssq[256];
    const int tid = threadIdx.x;
    long i = (long)blockIdx.x * 256 + tid;
    const long stride = (long)gridDim.x * 256;
    float s = 0.0f, q = 0.0f;
    for (; i < total; i += stride) {
        float v = h[i];
        s += v;
        q += v * v;
    }
    ssum[tid] = s;
    ssq[tid] = q;
    __syncthreads();
    if (tid < 128) {
        float S = ssum[tid] + ssum[tid + 128];
        float Q = ssq[tid] + ssq[tid + 128];
        unsafeAtomicAdd(&stats[tid], S);
        unsafeAtomicAdd(&stats[DIM + tid], Q);
    }
}

// ------------------------------------------------- BN + ReLU + residual
__global__ void bn_final_kernel(const float* __restrict__ h, const float* __restrict__ feat,
                                const float* __restrict__ stats, const float* __restrict__ gamma,
                                const float* __restrict__ beta, float* __restrict__ out, int n) {
    long idx = (long)blockIdx.x * blockDim.x + threadIdx.x;
    const long total = (long)n * DIM;
    if (idx >= total) return;
    const int c = (int)(idx & (DIM - 1));
    const float invN = 1.0f / (float)n;
    const float mean = stats[c] * invN;
    const float var = stats[DIM + c] * invN - mean * mean;
    const float inv = rsqrtf(var + BN_EPS);
    float y = (h[idx] - mean) * inv * gamma[c] + beta[c];
    y = fmaxf(y, 0.0f);
    out[idx] = feat[idx] + y;
}

// ---------------------------------------------------------------- launcher
extern "C" void kernel_launch(void* const* d_in, const int* in_sizes, int n_in,
                              void* d_out, int out_size, void* d_ws, size_t ws_size,
                              hipStream_t stream) {
    const float* feature = (const float*)d_in[0];
    const int* src       = (const int*)d_in[1];
    const int* dst       = (const int*)d_in[2];
    const float* W       = (const float*)d_in[3];
    const float* bias    = (const float*)d_in[4];
    const float* gamma   = (const float*)d_in[5];
    const float* beta    = (const float*)d_in[6];

    const int N = in_sizes[0] / DIM;
    const int E = in_sizes[1];

    // workspace layout (floats)
    float* ws = (float*)d_ws;
    float* out_deg = ws;                              // N
    float* in_deg  = ws + N;                          // N
    float* xbuf    = ws + 2L * N;                     // N*DIM   (x, later reused as h)
    float* agg     = ws + 2L * N + (long)N * DIM;     // N*DIM
    float* stats   = ws + 2L * N + 2L * N * DIM;      // 256
    float* hbuf    = xbuf;                            // x no longer needed after scatter

    // zero accumulation regions (re-zeroed every launch; graph-replay safe)
    {
        long zn = 2L * N;
        zero_kernel<<<(int)((zn + 255) / 256), 256, 0, stream>>>(out_deg, zn);
        long zn2 = (long)N * DIM + 256;
        int blocks = (int)((zn2 + 255) / 256);
        if (blocks > 16384) blocks = 16384;
        zero_kernel<<<blocks, 256, 0, stream>>>(agg, zn2);
    }

    degree_kernel<<<(E + 255) / 256, 256, 0, stream>>>(src, dst, out_deg, in_deg, E);

    {
        long t = (long)N * (DIM / 4);
        scale_src_kernel<<<(int)((t + 255) / 256), 256, 0, stream>>>(feature, out_deg, xbuf, N);
    }

    {
        long t = (long)E * 32;
        edge_scatter_kernel<<<(int)((t + 255) / 256), 256, 0, stream>>>(src, dst, xbuf, agg, E);
    }

    {
        int mtiles = (N + 15) / 16;
        int blocks = (mtiles + 3) / 4;
        gemm_wmma_kernel<<<blocks, 128, 0, stream>>>(agg, in_deg, W, bias, hbuf, N);
    }

    bn_stats_kernel<<<1024, 256, 0, stream>>>(hbuf, stats, (long)N * DIM);

    {
        long t = (long)N * DIM;
        bn_final_kernel<<<(int)((t + 255) / 256), 256, 0, stream>>>(
            hbuf, feature, stats, gamma, beta, (float*)d_out, N);
    }
}